// GraphConvolution_71098888618113
// MI455X (gfx1250) — compile-verified
//
#include <hip/hip_runtime.h>

// GraphConvolution for MI455X (gfx1250, wave32, WMMA).
// out = D (adj+I) D (att W), D = diag(rsqrt(rowsum(adj)))
// B=64, N=1024, DIN=DOUT=64.
//
// K3 (dominant GEMM) design: one wave owns a 16-row x 64-col output slab with
// 4 accumulators, so each adj A-fragment is loaded from HBM exactly once and
// feeds 4 v_wmma_f32_16x16x4_f32. adj is the only real HBM traffic
// (2 x 256 MB over K1+K3 -> ~22 us roofline at 23.3 TB/s); xs (16 MB) and
// W stay L2-resident.

typedef float v2f __attribute__((ext_vector_type(2)));
typedef float v8f __attribute__((ext_vector_type(8)));

#define BATCH 64
#define NN    1024
#define DD    64

// ---------------------------------------------------------------------------
// Kernel 1: dinv[row] = rsqrt(sum_j adj[row, j]); one wave32 per row.
// ---------------------------------------------------------------------------
__global__ __launch_bounds__(256) void gcn_deg_kernel(
    const float* __restrict__ adj, float* __restrict__ dinv) {
  const int wave = (blockIdx.x * blockDim.x + threadIdx.x) >> 5;  // global row
  const int lane = threadIdx.x & 31;
  const float* row = adj + (size_t)wave * NN;
  float s = 0.0f;
#pragma unroll
  for (int t = 0; t < 8; ++t) {  // 8 * (32 lanes * float4) = 1024 floats
    const float4 v = *(const float4*)(row + (size_t)lane * 4 + t * 128);
    s += (v.x + v.y) + (v.z + v.w);
  }
#pragma unroll
  for (int off = 16; off >= 1; off >>= 1) s += __shfl_xor(s, off, 32);
  if (lane == 0) dinv[wave] = rsqrtf(s);
}

// ---------------------------------------------------------------------------
// Kernel 2: xs[row, n] = dinv[row] * sum_k att[row, k] * W[k, n]
// One wave per 16-row x 64-col slab (4 accumulators, A-frag reused 4x).
// Block = 8 waves = 128 rows. Grid = B*N/128.
// ---------------------------------------------------------------------------
__global__ __launch_bounds__(256) void gcn_xw_kernel(
    const float* __restrict__ att, const float* __restrict__ W,
    const float* __restrict__ dinv, float* __restrict__ xs) {
  const int lane = threadIdx.x & 31;
  const int w    = threadIdx.x >> 5;   // wave in block: 0..7
  const int lm   = lane & 15;
  const int half = lane >> 4;          // 0: lanes 0-15, 1: lanes 16-31
  const int r0 = blockIdx.x * 128 + w * 16;        // global row tile

  v8f acc[4] = {{}, {}, {}, {}};
  const float* arow = att + (size_t)(r0 + lm) * DD + 2 * half;
  const float* bcol = W + (size_t)(2 * half) * DD + lm;
#pragma unroll
  for (int k = 0; k < DD; k += 4) {
    // A 16x4 frag: lane holds att[r0+lm, k+2h .. k+2h+1] (8B contiguous)
    v2f a = *(const v2f*)(arow + k);
#pragma unroll
    for (int t = 0; t < 4; ++t) {
      // B 4x16 frag for col tile t: W[k+2h, 16t+lm], W[k+2h+1, 16t+lm]
      v2f b;
      b.x = bcol[(size_t)k * DD + t * 16];
      b.y = bcol[(size_t)k * DD + DD + t * 16];
      acc[t] = __builtin_amdgcn_wmma_f32_16x16x4_f32(
          false, a, false, b, (short)0, acc[t], false, false);
    }
  }
#pragma unroll
  for (int r = 0; r < 8; ++r) {  // C/D layout: VGPR r -> row r (+8 hi half)
    const int i = r0 + r + 8 * half;
    const float dv = dinv[i];
#pragma unroll
    for (int t = 0; t < 4; ++t)
      xs[(size_t)i * DD + t * 16 + lm] = dv * acc[t][r];
  }
}

// ---------------------------------------------------------------------------
// Kernel 3: out[b,i,n] = dinv[b,i] * ( sum_j adj[b,i,j]*xs[b,j,n] + xs[b,i,n] )
// One wave per 16-row x 64-col slab of one batch; K loop = 1024.
// Each adj A-fragment (HBM stream) feeds 4 WMMAs; xs B-frags come from L2.
// Block = 8 waves = 128 rows. Grid = B * (N/128) = 512.
// ---------------------------------------------------------------------------
__global__ __launch_bounds__(256) void gcn_spmm_kernel(
    const float* __restrict__ adj, const float* __restrict__ xs,
    const float* __restrict__ dinv, float* __restrict__ out) {
  const int lane = threadIdx.x & 31;
  const int w    = threadIdx.x >> 5;
  const int lm   = lane & 15;
  const int half = lane >> 4;
  const int b  = blockIdx.x >> 3;              // 8 row-blocks per batch
  const int rb = blockIdx.x & 7;
  const int i0 = rb * 128 + w * 16;            // row tile within batch

  const float* adjb = adj + (size_t)b * NN * NN;
  const float* xsb  = xs + (size_t)b * NN * DD;

  v8f acc[4] = {{}, {}, {}, {}};
  const float* arow = adjb + (size_t)(i0 + lm) * NN + 2 * half;
  const float* bcol = xsb + (size_t)(2 * half) * DD + lm;

  for (int k0 = 0; k0 < NN; k0 += 32) {
    // Stream-prefetch the adj row ~1KB ahead (per-lane -> covers all 16 rows).
    __builtin_prefetch(arow + k0 + 256, 0, 0);
#pragma unroll
    for (int kk = 0; kk < 32; kk += 4) {
      const int k = k0 + kk;
      v2f a = *(const v2f*)(arow + k);         // adj[i0+lm, k+2h .. k+2h+1]
#pragma unroll
      for (int t = 0; t < 4; ++t) {
        v2f bf;
        bf.x = bcol[(size_t)k * DD + t * 16];        // xs[k+2h,   16t+lm]
        bf.y = bcol[(size_t)k * DD + DD + t * 16];   // xs[k+2h+1, 16t+lm]
        acc[t] = __builtin_amdgcn_wmma_f32_16x16x4_f32(
            false, a, false, bf, (short)0, acc[t], false, false);
      }
    }
  }
#pragma unroll
  for (int r = 0; r < 8; ++r) {
    const int i = i0 + r + 8 * half;
    const float dv = dinv[(size_t)b * NN + i];
#pragma unroll
    for (int t = 0; t < 4; ++t) {
      const int n = t * 16 + lm;
      const size_t gi = ((size_t)b * NN + i) * DD + n;
      // +I self-loop contributes xs[b,i,n]; then left D scaling.
      out[gi] = dv * (acc[t][r] + xsb[(size_t)i * DD + n]);
    }
  }
}

// ---------------------------------------------------------------------------
extern "C" void kernel_launch(void* const* d_in, const int* in_sizes, int n_in,
                              void* d_out, int out_size, void* d_ws, size_t ws_size,
                              hipStream_t stream) {
  const float* adj = (const float*)d_in[0];   // [B, N, N]
  const float* att = (const float*)d_in[1];   // [B, N, DIN]
  const float* Wt  = (const float*)d_in[2];   // [DIN, DOUT]
  float* out = (float*)d_out;                 // [B*N*DOUT] out, then [64*64] weight

  float* dinv = (float*)d_ws;                 // B*N floats       (256 KB)
  float* xs   = dinv + (size_t)BATCH * NN;    // B*N*DOUT floats  (16 MB)

  // K1: degrees -> dinv. One wave per row; 8 waves per block.
  gcn_deg_kernel<<<(BATCH * NN) / 8, 256, 0, stream>>>(adj, dinv);

  // K2: xs = dinv * (att @ W). 128 rows per block over B*N rows.
  gcn_xw_kernel<<<(BATCH * NN) / 128, 256, 0, stream>>>(att, Wt, dinv, xs);

  // K3: out = dinv * ((adj + I) @ xs). 128-row tiles, 8 per batch, 64 batches.
  gcn_spmm_kernel<<<BATCH * (NN / 128), 256, 0, stream>>>(adj, xs, dinv, out);

  // Second tuple element: passthrough weight, appended after out.
  hipMemcpyAsync(out + (size_t)BATCH * NN * DD, Wt,
                 (size_t)DD * DD * sizeof(float),
                 hipMemcpyDeviceToDevice, stream);
}